// SphereDownGeo_7361573946002
// MI455X (gfx1250) — compile-verified
//
#include <hip/hip_runtime.h>
#include <hip/hip_bf16.h>
#include <stdint.h>

// out[bc, i] = max over k of x[bc, children[i, k]],  bc in [0, B*C), i in [0, N_OUT)
// Pure bandwidth-bound gather+max: ~1.06 GB of traffic -> ~45us floor @ 23.3 TB/s.

typedef int __attribute__((ext_vector_type(4))) v4i;
// Address-space-qualified pointee types for the CDNA5 async-copy builtin:
typedef v4i __attribute__((address_space(1))) gv4i;   // global
typedef v4i __attribute__((address_space(3))) lv4i;   // LDS

#define TPB 256

__global__ __launch_bounds__(TPB) void sphere_down_max_kernel(
    const float* __restrict__ x,
    const int*   __restrict__ children,
    float*       __restrict__ out,
    int n_out, int n_in_per, int bc)
{
    __shared__ v4i s_idx[TPB];

    const int tid = threadIdx.x;
    const long long i = (long long)blockIdx.x * TPB + tid;
    const bool valid = (i < (long long)n_out);

    // ---- Stage this block's 4KB of children indices into LDS ----
    // CDNA5 async global->LDS copy (ASYNCcnt path); fallback is a plain
    // vector load + LDS store if the builtin is unavailable.
#if __has_builtin(__builtin_amdgcn_global_load_async_to_lds_b128)
    if (valid) {
        __builtin_amdgcn_global_load_async_to_lds_b128(
            (gv4i*)(children + i * 4),
            (lv4i*)(&s_idx[tid]),
            /*imm offset*/ 0, /*cpol*/ 0);
    }
#if __has_builtin(__builtin_amdgcn_s_wait_asynccnt)
    __builtin_amdgcn_s_wait_asynccnt(0);
#else
    asm volatile("s_wait_asynccnt 0" ::: "memory");
#endif
#else
    if (valid) {
        s_idx[tid] = *(const v4i*)(children + i * 4);
    }
#endif
    __syncthreads();

    if (!valid) return;
    const v4i c = s_idx[tid];

    // ---- 16 (b,c) maps share the same gather indices ----
    // Streams are far larger than the 192MB L2 with zero reuse: use
    // non-temporal loads/stores so we don't rinse the cache.
#pragma unroll 4
    for (int b = 0; b < bc; ++b) {
        const float* xb = x + (size_t)b * (size_t)n_in_per;
        float v0 = __builtin_nontemporal_load(xb + c.x);
        float v1 = __builtin_nontemporal_load(xb + c.y);
        float v2 = __builtin_nontemporal_load(xb + c.z);
        float v3 = __builtin_nontemporal_load(xb + c.w);
        float m  = fmaxf(fmaxf(v0, v1), fmaxf(v2, v3));
        __builtin_nontemporal_store(m, out + (size_t)b * (size_t)n_out + i);
    }
}

extern "C" void kernel_launch(void* const* d_in, const int* in_sizes, int n_in,
                              void* d_out, int out_size, void* d_ws, size_t ws_size,
                              hipStream_t stream) {
    const float* x        = (const float*)d_in[0];   // (B, C, N_IN) fp32
    const int*   children = (const int*)d_in[1];     // (N_OUT, 4) int32
    float*       out      = (float*)d_out;           // (B, C, N_OUT) fp32

    const int n_children = in_sizes[1];      // N_OUT * 4
    const int n_out_per  = n_children / 4;   // N_OUT
    const int bc         = out_size / n_out_per;      // B*C (= 16)
    const int n_in_per   = in_sizes[0] / bc;          // N_IN

    const int grid = (n_out_per + TPB - 1) / TPB;
    sphere_down_max_kernel<<<grid, TPB, 0, stream>>>(x, children, out,
                                                     n_out_per, n_in_per, bc);
}